// DGCNNAlt_47193100648616
// MI455X (gfx1250) — compile-verified
//
#include <hip/hip_runtime.h>
#include <stdint.h>

#define BATCH 32
#define NPTS  1024
#define KNN   20
#define EDGES (BATCH * NPTS * KNN)   // 655360
#define EPSBN 1e-5f
#define INV_E (1.0f / 655360.0f)
#define ALD   65                     // padded LDS row stride (bank-conflict free)

typedef float v2f __attribute__((ext_vector_type(2)));
typedef float v8f __attribute__((ext_vector_type(8)));

__device__ __forceinline__ v8f wmma_step(v8f acc, v2f a, v2f b) {
  return __builtin_amdgcn_wmma_f32_16x16x4_f32(false, a, false, b,
                                               (short)0, acc, false, false);
}

// ---------------- ordered float <-> uint encoding (for atomic max) ----------
__device__ __forceinline__ unsigned f2ord(float f) {
  unsigned u = __float_as_uint(f);
  return (u & 0x80000000u) ? ~u : (u | 0x80000000u);
}
__device__ __forceinline__ float ord2f(unsigned e) {
  unsigned u = (e & 0x80000000u) ? (e & 0x7fffffffu) : ~e;
  return __uint_as_float(u);
}
#define ORD_NEG_INF 0x007fffffu   // f2ord(-inf)

// ---------------- top-k (k=20) sorted insertion, stays in registers --------
__device__ __forceinline__ void topk_insert(float (&bd)[KNN], int (&bi)[KNN],
                                            float d, int j) {
  if (d < bd[KNN - 1]) {
    bd[KNN - 1] = d; bi[KNN - 1] = j;
#pragma unroll
    for (int t = KNN - 1; t > 0; --t) {
      if (bd[t] < bd[t - 1]) {
        float td = bd[t]; bd[t] = bd[t - 1]; bd[t - 1] = td;
        int   ti = bi[t]; bi[t] = bi[t - 1]; bi[t - 1] = ti;
      }
    }
  }
}

// ======================= kernel 0: init ====================================
__global__ void k_init(float* acc1, float* acc2, unsigned* pooled) {
  int t = blockIdx.x * 256 + threadIdx.x;
  if (t < 128) { acc1[t] = 0.0f; acc2[t] = 0.0f; }
  if (t < BATCH * NPTS) pooled[t] = ORD_NEG_INF;
}

// ======================= kernel 1: kNN in coordinate space (D=3) ===========
__global__ void k_knn3(const float* __restrict__ pos, int* __restrict__ idx1) {
  __shared__ float sp[NPTS * 3];
  __shared__ float sq[NPTS];
  int b = blockIdx.x >> 2;
  int tile = blockIdx.x & 3;
  const float* pb = pos + (size_t)b * NPTS * 3;
  for (int t = threadIdx.x; t < NPTS * 3; t += 256) sp[t] = pb[t];
  __syncthreads();
  for (int t = threadIdx.x; t < NPTS; t += 256) {
    float x = sp[t * 3], y = sp[t * 3 + 1], z = sp[t * 3 + 2];
    sq[t] = x * x + y * y + z * z;
  }
  __syncthreads();
  int i = tile * 256 + threadIdx.x;
  float xi = sp[i * 3], yi = sp[i * 3 + 1], zi = sp[i * 3 + 2];
  float ni = sq[i];
  float bd[KNN]; int bi[KNN];
#pragma unroll
  for (int t = 0; t < KNN; ++t) { bd[t] = 3.4e38f; bi[t] = 0; }
  for (int j = 0; j < NPTS; ++j) {
    float dot = xi * sp[j * 3] + yi * sp[j * 3 + 1] + zi * sp[j * 3 + 2];
    float d = ni + sq[j] - 2.0f * dot;
    topk_insert(bd, bi, d, j);
  }
  int* o = idx1 + ((size_t)b * NPTS + i) * KNN;
#pragma unroll
  for (int t = 0; t < KNN; ++t) o[t] = bi[t];
}

// ===== EdgeConv1, WMMA tiling: block = 160 edges (10 M-tiles), N=64 ========
// layer1: K=8 (6 padded to 8), layers 2/3: K=64.

// ---- kernel 2: BN stats for layer 1 (raw h1 = e@W1+b1) --------------------
__global__ void k_conv1_stats1(const float* __restrict__ pos,
                               const int* __restrict__ idx1,
                               const float* __restrict__ W1,
                               const float* __restrict__ b1, float* acc1) {
  __shared__ float sp[NPTS * 3];
  __shared__ float sef[160 * 8];
  __shared__ float sW1[8 * 64];
  __shared__ float sb1[64];
  __shared__ float ls[64], lq[64];
  int e0 = blockIdx.x * 160;
  int b = e0 / (NPTS * KNN);
  int tid = threadIdx.x;
  const float* pb = pos + (size_t)b * NPTS * 3;
  for (int t = tid; t < NPTS * 3; t += 256) sp[t] = pb[t];
  for (int t = tid; t < 8 * 64; t += 256) sW1[t] = (t < 6 * 64) ? W1[t] : 0.0f;
  if (tid < 64) { sb1[tid] = b1[tid]; ls[tid] = 0.0f; lq[tid] = 0.0f; }
  __syncthreads();
  if (tid < 160) {
    int e = e0 + tid;
    int rem = e - b * NPTS * KNN;
    int i = rem / KNN;
    int j = idx1[e];
    float xi = sp[i * 3], yi = sp[i * 3 + 1], zi = sp[i * 3 + 2];
    float* r = sef + tid * 8;
    r[0] = xi; r[1] = yi; r[2] = zi;
    r[3] = sp[j * 3] - xi; r[4] = sp[j * 3 + 1] - yi; r[5] = sp[j * 3 + 2] - zi;
    r[6] = 0.0f; r[7] = 0.0f;
  }
  __syncthreads();
  int wave = tid >> 5, lane = tid & 31, row = lane & 15, half = lane >> 4;
  for (int jj = 0; jj < 5; ++jj) {
    int job = jj * 8 + wave;          // 0..39 : 10 mt x 4 nt
    int mt = job >> 2, nt = job & 3;
    v8f acc = {};
    const float* Ar = sef + (mt * 16 + row) * 8;
#pragma unroll
    for (int kt = 0; kt < 2; ++kt) {
      int k0 = kt * 4 + half * 2;
      v2f af = { Ar[k0], Ar[k0 + 1] };
      v2f bf = { sW1[k0 * 64 + nt * 16 + row], sW1[(k0 + 1) * 64 + nt * 16 + row] };
      acc = wmma_step(acc, af, bf);
    }
    float bb = sb1[nt * 16 + row];
    float s = 0.0f, q = 0.0f;
#pragma unroll
    for (int v = 0; v < 8; ++v) { float h = acc[v] + bb; s += h; q += h * h; }
    s += __shfl_xor(s, 16, 32);
    q += __shfl_xor(q, 16, 32);
    if (half == 0) { atomicAdd(&ls[nt * 16 + row], s); atomicAdd(&lq[nt * 16 + row], q); }
  }
  __syncthreads();
  if (tid < 64) {
    atomicAdd(&acc1[tid], ls[tid]);
    atomicAdd(&acc1[64 + tid], lq[tid]);
  }
}

// ---- kernel 3: BN stats for layer 2 ---------------------------------------
__global__ void k_conv1_stats2(const float* __restrict__ pos,
                               const int* __restrict__ idx1,
                               const float* __restrict__ W1, const float* __restrict__ b1,
                               const float* __restrict__ g1, const float* __restrict__ be1,
                               const float* __restrict__ acc1,
                               const float* __restrict__ W2, const float* __restrict__ b2,
                               float* acc2) {
  extern __shared__ float sm2[];
  float* sp   = sm2;                  // 3072
  float* sef  = sp + 3072;            // 160*8
  float* sW1  = sef + 160 * 8;        // 512
  float* sW2  = sW1 + 512;            // 4096
  float* act1 = sW2 + 4096;           // 160*ALD
  float* sb1  = act1 + 160 * ALD;     // 64
  float* sb2  = sb1 + 64;             // 64
  float* sk1  = sb2 + 64;             // 64
  float* sh1  = sk1 + 64;             // 64
  float* ls   = sh1 + 64;             // 64
  float* lq   = ls + 64;              // 64
  int e0 = blockIdx.x * 160;
  int b = e0 / (NPTS * KNN);
  int tid = threadIdx.x;
  const float* pb = pos + (size_t)b * NPTS * 3;
  for (int t = tid; t < NPTS * 3; t += 256) sp[t] = pb[t];
  for (int t = tid; t < 8 * 64; t += 256) sW1[t] = (t < 6 * 64) ? W1[t] : 0.0f;
  for (int t = tid; t < 64 * 64; t += 256) sW2[t] = W2[t];
  if (tid < 64) {
    int c = tid;
    sb1[c] = b1[c]; sb2[c] = b2[c];
    float mean = acc1[c] * INV_E;
    float var = acc1[64 + c] * INV_E - mean * mean;
    float kf = g1[c] * rsqrtf(var + EPSBN);
    sk1[c] = kf; sh1[c] = be1[c] - kf * mean;
    ls[c] = 0.0f; lq[c] = 0.0f;
  }
  __syncthreads();
  if (tid < 160) {
    int e = e0 + tid;
    int rem = e - b * NPTS * KNN;
    int i = rem / KNN;
    int j = idx1[e];
    float xi = sp[i * 3], yi = sp[i * 3 + 1], zi = sp[i * 3 + 2];
    float* r = sef + tid * 8;
    r[0] = xi; r[1] = yi; r[2] = zi;
    r[3] = sp[j * 3] - xi; r[4] = sp[j * 3 + 1] - yi; r[5] = sp[j * 3 + 2] - zi;
    r[6] = 0.0f; r[7] = 0.0f;
  }
  __syncthreads();
  int wave = tid >> 5, lane = tid & 31, row = lane & 15, half = lane >> 4;
  // phase 1: h1 -> bn -> relu -> act1
  for (int jj = 0; jj < 5; ++jj) {
    int job = jj * 8 + wave; int mt = job >> 2, nt = job & 3;
    v8f acc = {};
    const float* Ar = sef + (mt * 16 + row) * 8;
#pragma unroll
    for (int kt = 0; kt < 2; ++kt) {
      int k0 = kt * 4 + half * 2;
      v2f af = { Ar[k0], Ar[k0 + 1] };
      v2f bf = { sW1[k0 * 64 + nt * 16 + row], sW1[(k0 + 1) * 64 + nt * 16 + row] };
      acc = wmma_step(acc, af, bf);
    }
    int col = nt * 16 + row;
    float bb = sb1[col], kf = sk1[col], sh = sh1[col];
#pragma unroll
    for (int v = 0; v < 8; ++v)
      act1[(mt * 16 + v + half * 8) * ALD + col] = fmaxf(kf * (acc[v] + bb) + sh, 0.0f);
  }
  __syncthreads();
  // phase 2: h2 = act1 @ W2 + b2 ; accumulate stats
  for (int jj = 0; jj < 5; ++jj) {
    int job = jj * 8 + wave; int mt = job >> 2, nt = job & 3;
    v8f acc = {};
    const float* Ar = act1 + (mt * 16 + row) * ALD;
#pragma unroll
    for (int kt = 0; kt < 16; ++kt) {
      int k0 = kt * 4 + half * 2;
      v2f af = { Ar[k0], Ar[k0 + 1] };
      v2f bf = { sW2[k0 * 64 + nt * 16 + row], sW2[(k0 + 1) * 64 + nt * 16 + row] };
      acc = wmma_step(acc, af, bf);
    }
    float bb = sb2[nt * 16 + row];
    float s = 0.0f, q = 0.0f;
#pragma unroll
    for (int v = 0; v < 8; ++v) { float h = acc[v] + bb; s += h; q += h * h; }
    s += __shfl_xor(s, 16, 32);
    q += __shfl_xor(q, 16, 32);
    if (half == 0) { atomicAdd(&ls[nt * 16 + row], s); atomicAdd(&lq[nt * 16 + row], q); }
  }
  __syncthreads();
  if (tid < 64) {
    atomicAdd(&acc2[tid], ls[tid]);
    atomicAdd(&acc2[64 + tid], lq[tid]);
  }
}

// ---- kernel 4: EdgeConv1 final: 3 WMMA layers + fused max over k ----------
// block = 8 nodes (160 edges), grid = B*N/8 = 4096
__global__ void k_conv1_final(const float* __restrict__ pos,
                              const int* __restrict__ idx1,
                              const float* __restrict__ W1, const float* __restrict__ b1,
                              const float* __restrict__ g1, const float* __restrict__ be1,
                              const float* __restrict__ acc1,
                              const float* __restrict__ W2, const float* __restrict__ b2,
                              const float* __restrict__ g2, const float* __restrict__ be2,
                              const float* __restrict__ acc2,
                              const float* __restrict__ W3, const float* __restrict__ b3,
                              float* __restrict__ x1) {
  extern __shared__ float smf[];
  float* sp   = smf;                  // 3072
  float* sef  = sp + 3072;            // 1280
  float* sW1  = sef + 1280;           // 512
  float* sW2  = sW1 + 512;            // 4096
  float* sW3  = sW2 + 4096;           // 4096
  float* act1 = sW3 + 4096;           // 160*ALD (reused for h3)
  float* act2 = act1 + 160 * ALD;     // 160*ALD
  float* sb1  = act2 + 160 * ALD;     // 7 x 64
  float* sb2  = sb1 + 64;
  float* sb3  = sb2 + 64;
  float* sk1  = sb3 + 64;
  float* sh1  = sk1 + 64;
  float* sk2  = sh1 + 64;
  float* sh2  = sk2 + 64;
  int b = blockIdx.x >> 7;
  int n0 = (blockIdx.x & 127) * 8;
  int tid = threadIdx.x;
  const float* pb = pos + (size_t)b * NPTS * 3;
  for (int t = tid; t < NPTS * 3; t += 256) sp[t] = pb[t];
  for (int t = tid; t < 8 * 64; t += 256) sW1[t] = (t < 6 * 64) ? W1[t] : 0.0f;
  for (int t = tid; t < 64 * 64; t += 256) { sW2[t] = W2[t]; sW3[t] = W3[t]; }
  if (tid < 64) {
    int c = tid;
    sb1[c] = b1[c]; sb2[c] = b2[c]; sb3[c] = b3[c];
    float m1 = acc1[c] * INV_E;
    float v1 = acc1[64 + c] * INV_E - m1 * m1;
    float k1 = g1[c] * rsqrtf(v1 + EPSBN);
    sk1[c] = k1; sh1[c] = be1[c] - k1 * m1;
    float m2 = acc2[c] * INV_E;
    float v2 = acc2[64 + c] * INV_E - m2 * m2;
    float k2 = g2[c] * rsqrtf(v2 + EPSBN);
    sk2[c] = k2; sh2[c] = be2[c] - k2 * m2;
  }
  __syncthreads();
  if (tid < 160) {
    int n = tid / KNN, kk = tid % KNN;
    int i = n0 + n;
    int j = idx1[((size_t)b * NPTS + i) * KNN + kk];
    float xi = sp[i * 3], yi = sp[i * 3 + 1], zi = sp[i * 3 + 2];
    float* r = sef + tid * 8;
    r[0] = xi; r[1] = yi; r[2] = zi;
    r[3] = sp[j * 3] - xi; r[4] = sp[j * 3 + 1] - yi; r[5] = sp[j * 3 + 2] - zi;
    r[6] = 0.0f; r[7] = 0.0f;
  }
  __syncthreads();
  int wave = tid >> 5, lane = tid & 31, row = lane & 15, half = lane >> 4;
  // phase 1: layer1 -> bn1 -> relu -> act1
  for (int jj = 0; jj < 5; ++jj) {
    int job = jj * 8 + wave; int mt = job >> 2, nt = job & 3;
    v8f acc = {};
    const float* Ar = sef + (mt * 16 + row) * 8;
#pragma unroll
    for (int kt = 0; kt < 2; ++kt) {
      int k0 = kt * 4 + half * 2;
      v2f af = { Ar[k0], Ar[k0 + 1] };
      v2f bf = { sW1[k0 * 64 + nt * 16 + row], sW1[(k0 + 1) * 64 + nt * 16 + row] };
      acc = wmma_step(acc, af, bf);
    }
    int col = nt * 16 + row;
    float bb = sb1[col], kf = sk1[col], sh = sh1[col];
#pragma unroll
    for (int v = 0; v < 8; ++v)
      act1[(mt * 16 + v + half * 8) * ALD + col] = fmaxf(kf * (acc[v] + bb) + sh, 0.0f);
  }
  __syncthreads();
  // phase 2: layer2 -> bn2 -> relu -> act2
  for (int jj = 0; jj < 5; ++jj) {
    int job = jj * 8 + wave; int mt = job >> 2, nt = job & 3;
    v8f acc = {};
    const float* Ar = act1 + (mt * 16 + row) * ALD;
#pragma unroll
    for (int kt = 0; kt < 16; ++kt) {
      int k0 = kt * 4 + half * 2;
      v2f af = { Ar[k0], Ar[k0 + 1] };
      v2f bf = { sW2[k0 * 64 + nt * 16 + row], sW2[(k0 + 1) * 64 + nt * 16 + row] };
      acc = wmma_step(acc, af, bf);
    }
    int col = nt * 16 + row;
    float bb = sb2[col], kf = sk2[col], sh = sh2[col];
#pragma unroll
    for (int v = 0; v < 8; ++v)
      act2[(mt * 16 + v + half * 8) * ALD + col] = fmaxf(kf * (acc[v] + bb) + sh, 0.0f);
  }
  __syncthreads();
  // phase 3: layer3 (plain) -> act1 (reused as h3 buffer)
  for (int jj = 0; jj < 5; ++jj) {
    int job = jj * 8 + wave; int mt = job >> 2, nt = job & 3;
    v8f acc = {};
    const float* Ar = act2 + (mt * 16 + row) * ALD;
#pragma unroll
    for (int kt = 0; kt < 16; ++kt) {
      int k0 = kt * 4 + half * 2;
      v2f af = { Ar[k0], Ar[k0 + 1] };
      v2f bf = { sW3[k0 * 64 + nt * 16 + row], sW3[(k0 + 1) * 64 + nt * 16 + row] };
      acc = wmma_step(acc, af, bf);
    }
    int col = nt * 16 + row;
    float bb = sb3[col];
#pragma unroll
    for (int v = 0; v < 8; ++v)
      act1[(mt * 16 + v + half * 8) * ALD + col] = acc[v] + bb;
  }
  __syncthreads();
  // max over k=20 edges per node
  for (int t = tid; t < 8 * 64; t += 256) {
    int n = t >> 6, c = t & 63;
    float m = act1[(n * KNN) * ALD + c];
    for (int kk = 1; kk < KNN; ++kk) m = fmaxf(m, act1[(n * KNN + kk) * ALD + c]);
    x1[((size_t)b * NPTS + n0 + n) * 64 + c] = m;
  }
}

// ======================= kernel 5: kNN in feature space (WMMA Gram) ========
// block = 256 query nodes; j processed in chunks of 64 (G tile in LDS)
__global__ void k_knn_feat(const float* __restrict__ x1, int* __restrict__ idx2) {
  extern __shared__ float smk[];
  float* sxi = smk;                   // 256*ALD
  float* sxj = sxi + 256 * ALD;       // 64*ALD
  float* G   = sxj + 64 * ALD;        // 256*64
  float* sn  = G + 256 * 64;          // 1024
  int b = blockIdx.x >> 2;
  int tile = blockIdx.x & 3;
  int tid = threadIdx.x;
  const float* xb = x1 + (size_t)b * NPTS * 64;
  for (int t = tid; t < NPTS; t += 256) {
    float s = 0.0f;
    for (int d = 0; d < 64; ++d) { float v = xb[t * 64 + d]; s += v * v; }
    sn[t] = s;
  }
  for (int t = tid; t < 256 * 64; t += 256) {
    int r = t >> 6, c = t & 63;
    sxi[r * ALD + c] = xb[(tile * 256 + r) * 64 + c];
  }
  float bd[KNN]; int bi[KNN];
#pragma unroll
  for (int t = 0; t < KNN; ++t) { bd[t] = 3.4e38f; bi[t] = 0; }
  __syncthreads();
  float ni = sn[tile * 256 + tid];
  int wave = tid >> 5, lane = tid & 31, row = lane & 15, half = lane >> 4;
  for (int jc = 0; jc < 16; ++jc) {
    __syncthreads();  // protect sxj/G from previous iteration readers
    for (int t = tid; t < 64 * 64; t += 256) {
      int r = t >> 6, c = t & 63;
      sxj[r * ALD + c] = xb[(jc * 64 + r) * 64 + c];
    }
    __syncthreads();
    // G[256x64] = Xi @ Xj^T : 16 mt x 4 nt = 64 jobs / 8 waves
    for (int jj = 0; jj < 8; ++jj) {
      int job = jj * 8 + wave; int mt = job >> 2, nt = job & 3;
      v8f acc = {};
      const float* Ar = sxi + (mt * 16 + row) * ALD;
      const float* Br = sxj + (nt * 16 + row) * ALD;  // col-major access: B[k][n]=xj[n][k]
#pragma unroll
      for (int kt = 0; kt < 16; ++kt) {
        int k0 = kt * 4 + half * 2;
        v2f af = { Ar[k0], Ar[k0 + 1] };
        v2f bf = { Br[k0], Br[k0 + 1] };
        acc = wmma_step(acc, af, bf);
      }
#pragma unroll
      for (int v = 0; v < 8; ++v)
        G[(mt * 16 + v + half * 8) * 64 + nt * 16 + row] = acc[v];
    }
    __syncthreads();
    const float* Gr = G + tid * 64;
    for (int jj = 0; jj < 64; ++jj) {
      int j = jc * 64 + jj;
      float dd = ni + sn[j] - 2.0f * Gr[jj];
      topk_insert(bd, bi, dd, j);
    }
  }
  int* o = idx2 + ((size_t)b * NPTS + tile * 256 + tid) * KNN;
#pragma unroll
  for (int t = 0; t < KNN; ++t) o[t] = bi[t];
}

// ======================= kernel 6: EdgeConv2 (WMMA) ========================
// block = 4 nodes (80 edges, 5 M-tiles), N=128 (8 tiles), K=128 (32 steps)
__global__ void k_conv2(const float* __restrict__ x1, const int* __restrict__ idx2,
                        const float* __restrict__ Wc2, const float* __restrict__ bc2,
                        float* __restrict__ x2) {
  extern __shared__ float smem[];
  float* sA = smem;                 // 80*128
  float* sW = sA + 80 * 128;        // 128*128
  float* sO = sW + 128 * 128;       // 80*128
  float* sb = sO + 80 * 128;        // 128
  int b = blockIdx.x >> 8;          // 256 blocks per batch
  int n0 = (blockIdx.x & 255) * 4;
  int tid = threadIdx.x;
  for (int t = tid; t < 128 * 128; t += 256) sW[t] = Wc2[t];
  if (tid < 128) sb[tid] = bc2[tid];
  const float* xb = x1 + (size_t)b * NPTS * 64;
  for (int t = tid; t < 80 * 128; t += 256) {
    int r = t >> 7, c = t & 127;
    int n = r / KNN, kk = r % KNN;
    int i = n0 + n;
    int j = idx2[((size_t)b * NPTS + i) * KNN + kk];
    float v = (c < 64) ? xb[i * 64 + c]
                       : (xb[j * 64 + (c - 64)] - xb[i * 64 + (c - 64)]);
    sA[t] = v;
  }
  __syncthreads();
  int wave = tid >> 5, lane = tid & 31;
  int row = lane & 15, half = lane >> 4;
  for (int jj = 0; jj < 5; ++jj) {
    int job = jj * 8 + wave;        // 0..39
    int mt = job >> 3, nt = job & 7;
    v8f acc = {};
    const float* Arow = sA + (mt * 16 + row) * 128;
#pragma unroll
    for (int kt = 0; kt < 32; ++kt) {
      int k0 = kt * 4 + half * 2;
      v2f af = { Arow[k0], Arow[k0 + 1] };
      v2f bf = { sW[k0 * 128 + nt * 16 + row], sW[(k0 + 1) * 128 + nt * 16 + row] };
      acc = wmma_step(acc, af, bf);
    }
    float bb = sb[nt * 16 + row];   // col = lane&15
#pragma unroll
    for (int v = 0; v < 8; ++v) acc[v] += bb;
#pragma unroll
    for (int v = 0; v < 8; ++v)
      sO[(mt * 16 + v + half * 8) * 128 + nt * 16 + row] = acc[v];
  }
  __syncthreads();
  for (int t = tid; t < 4 * 128; t += 256) {
    int n = t >> 7, c = t & 127;
    float m = sO[(n * KNN) * 128 + c];
    for (int kk = 1; kk < KNN; ++kk) m = fmaxf(m, sO[(n * KNN + kk) * 128 + c]);
    x2[((size_t)b * NPTS + n0 + n) * 128 + c] = m;
  }
}

// ======================= kernel 7: lin (192->1024) + fused global max pool =
__global__ void k_lin_pool(const float* __restrict__ x1, const float* __restrict__ x2,
                           const float* __restrict__ Wl, const float* __restrict__ bl,
                           unsigned* __restrict__ pooled) {
  __shared__ float sA[16 * 192];
  int b = blockIdx.x >> 6;          // 64 blocks per batch
  int n0 = (blockIdx.x & 63) * 16;
  int tid = threadIdx.x;
  for (int t = tid; t < 16 * 192; t += 256) {
    int r = t / 192, c = t - r * 192;
    int i = n0 + r;
    float v = (c < 64) ? x1[((size_t)b * NPTS + i) * 64 + c]
                       : x2[((size_t)b * NPTS + i) * 128 + (c - 64)];
    sA[t] = v;
  }
  __syncthreads();
  int wave = tid >> 5, lane = tid & 31;
  int row = lane & 15, half = lane >> 4;
  const float* Arow = sA + row * 192;
  for (int jj = 0; jj < 8; ++jj) {
    int nt = jj * 8 + wave;         // 0..63
    int colg = nt * 16 + row;
    v8f acc = {};
#pragma unroll
    for (int kt = 0; kt < 48; ++kt) {
      int k0 = kt * 4 + half * 2;
      v2f af = { Arow[k0], Arow[k0 + 1] };
      v2f bf = { Wl[(size_t)k0 * 1024 + colg], Wl[(size_t)(k0 + 1) * 1024 + colg] };
      acc = wmma_step(acc, af, bf);
    }
    float bb = bl[colg];
    float mx = acc[0] + bb;
#pragma unroll
    for (int v = 1; v < 8; ++v) mx = fmaxf(mx, acc[v] + bb);
    float other = __shfl_xor(mx, 16, 32);
    mx = fmaxf(mx, other);
    if (half == 0) atomicMax(&pooled[b * 1024 + colg], f2ord(mx));
  }
}

// ======================= kernel 8: classification head =====================
__global__ void k_head(const unsigned* __restrict__ pooled,
                       const float* __restrict__ Wm1, const float* __restrict__ bm1,
                       const float* __restrict__ Wm2, const float* __restrict__ bm2,
                       const float* __restrict__ Wm3, const float* __restrict__ bm3,
                       float* __restrict__ out) {
  __shared__ float p[1024];
  __shared__ float h1[512];
  __shared__ float h2[256];
  int b = blockIdx.x, tid = threadIdx.x;
  for (int t = tid; t < 1024; t += 256) p[t] = ord2f(pooled[b * 1024 + t]);
  __syncthreads();
  for (int o = tid; o < 512; o += 256) {
    float s = bm1[o];
    for (int d = 0; d < 1024; ++d) s += p[d] * Wm1[d * 512 + o];
    h1[o] = fmaxf(s, 0.0f);
  }
  __syncthreads();
  if (tid < 256) {
    float s = bm2[tid];
    for (int d = 0; d < 512; ++d) s += h1[d] * Wm2[d * 256 + tid];
    h2[tid] = fmaxf(s, 0.0f);
  }
  __syncthreads();
  if (tid < 40) {
    float s = bm3[tid];
    for (int d = 0; d < 256; ++d) s += h2[d] * Wm3[d * 40 + tid];
    out[b * 40 + tid] = s;
  }
}

// ======================= launcher ==========================================
extern "C" void kernel_launch(void* const* d_in, const int* in_sizes, int n_in,
                              void* d_out, int out_size, void* d_ws, size_t ws_size,
                              hipStream_t stream) {
  (void)in_sizes; (void)n_in; (void)out_size; (void)ws_size;
  const float* pos = (const float*)d_in[0];
  const float* W1  = (const float*)d_in[1];
  const float* b1  = (const float*)d_in[2];
  const float* g1  = (const float*)d_in[3];
  const float* be1 = (const float*)d_in[4];
  const float* W2  = (const float*)d_in[5];
  const float* b2  = (const float*)d_in[6];
  const float* g2  = (const float*)d_in[7];
  const float* be2 = (const float*)d_in[8];
  const float* W3  = (const float*)d_in[9];
  const float* b3  = (const float*)d_in[10];
  const float* Wc2 = (const float*)d_in[11];
  const float* bc2 = (const float*)d_in[12];
  const float* Wl  = (const float*)d_in[13];
  const float* bl  = (const float*)d_in[14];
  const float* Wm1 = (const float*)d_in[15];
  const float* bm1 = (const float*)d_in[16];
  const float* Wm2 = (const float*)d_in[17];
  const float* bm2 = (const float*)d_in[18];
  const float* Wm3 = (const float*)d_in[19];
  const float* bm3 = (const float*)d_in[20];

  char* ws = (char*)d_ws;
  int*      idx1   = (int*)     (ws + 0);          // 2,621,440 B
  float*    x1     = (float*)   (ws + 2621440);    // 8,388,608 B
  int*      idx2   = (int*)     (ws + 11010048);   // 2,621,440 B
  float*    x2     = (float*)   (ws + 13631488);   // 16,777,216 B
  float*    acc1   = (float*)   (ws + 30408704);   // 512 B
  float*    acc2   = (float*)   (ws + 30409216);   // 512 B
  unsigned* pooled = (unsigned*)(ws + 30409728);   // 131,072 B

  size_t smem_s2 = (size_t)(3072 + 1280 + 512 + 4096 + 160 * ALD + 6 * 64) * 4;
  size_t smem_cf = (size_t)(3072 + 1280 + 512 + 4096 + 4096 + 2 * 160 * ALD + 7 * 64) * 4;
  size_t smem_kf = (size_t)(256 * ALD + 64 * ALD + 256 * 64 + 1024) * 4;
  size_t smem_c2 = (size_t)(80 * 128 + 128 * 128 + 80 * 128 + 128) * 4;

  k_init        <<<128,  256, 0, stream>>>(acc1, acc2, pooled);
  k_knn3        <<<128,  256, 0, stream>>>(pos, idx1);
  k_conv1_stats1<<<4096, 256, 0, stream>>>(pos, idx1, W1, b1, acc1);
  k_conv1_stats2<<<4096, 256, smem_s2, stream>>>(pos, idx1, W1, b1, g1, be1, acc1,
                                                 W2, b2, acc2);
  k_conv1_final <<<4096, 256, smem_cf, stream>>>(pos, idx1, W1, b1, g1, be1, acc1,
                                                 W2, b2, g2, be2, acc2, W3, b3, x1);
  k_knn_feat    <<<128,  256, smem_kf, stream>>>(x1, idx2);
  k_conv2       <<<8192, 256, smem_c2, stream>>>(x1, idx2, Wc2, bc2, x2);
  k_lin_pool    <<<2048, 256, 0, stream>>>(x1, x2, Wl, bl, pooled);
  k_head        <<<32,   256, 0, stream>>>(pooled, Wm1, bm1, Wm2, bm2, Wm3, bm3,
                                           (float*)d_out);
}